// IntegrationMeasure_5007931867607
// MI455X (gfx1250) — compile-verified
//
#include <hip/hip_runtime.h>
#include <hip/hip_bf16.h>

typedef float v2f __attribute__((ext_vector_type(2)));
typedef float v8f __attribute__((ext_vector_type(8)));

#define NUM_BINS 10
#define HSIZE (8 * NUM_BINS * NUM_BINS)   // 800 ints

// ---------------------------------------------------------------------------
// Per-wave 16-row tile of masked sums via V_WMMA_F32_16X16X4_F32.
// A: 16x4 chunk of states (f32).  Lane layout (ISA 7.12.2, 32-bit A 16x4):
//   lanes 0-15  : row = lane,    VGPR0=K0, VGPR1=K1
//   lanes 16-31 : row = lane-16, VGPR0=K2, VGPR1=K3
// B: 4x16 masks. Mirror layout: lanes 0-15 hold N=lane of K0/K1,
//   lanes 16-31 hold N=lane-16 of K2/K3.
// D: 16x16 f32, VGPR r: lanes 0-15 -> M=r, N=lane; lanes 16-31 -> M=r+8.
// Columns: 0-7 = partition masks, 8 = all-ones (row total), 9-15 = zero.
// ---------------------------------------------------------------------------
__device__ __forceinline__ v8f tile_sums(const float* __restrict__ states,
                                         long long tile, int rl, int h,
                                         const v2f bf[16]) {
  v8f acc = {0.f, 0.f, 0.f, 0.f, 0.f, 0.f, 0.f, 0.f};
  const float* base = states + (tile * 16 + rl) * 64 + 2 * h;
#pragma unroll
  for (int ck = 0; ck < 16; ++ck) {
    v2f a = *(const v2f*)(base + 4 * ck);   // 8B-aligned global_load_b64
    acc = __builtin_amdgcn_wmma_f32_16x16x4_f32(
        /*neg_a=*/false, a, /*neg_b=*/false, bf[ck],
        /*c_mod=*/(short)0, acc, /*reuse_a=*/false, /*reuse_b=*/false);
  }
  return acc;
}

__device__ __forceinline__ void build_bfrag(const unsigned char* __restrict__ parts,
                                            int rl, int h, v2f bf[16]) {
#pragma unroll
  for (int ck = 0; ck < 16; ++ck) {
    int d0 = 4 * ck + 2 * h;
    float m0, m1;
    if (rl < 8) {
      m0 = parts[rl * 64 + d0]     ? 1.f : 0.f;
      m1 = parts[rl * 64 + d0 + 1] ? 1.f : 0.f;
    } else if (rl == 8) {
      m0 = 1.f; m1 = 1.f;
    } else {
      m0 = 0.f; m1 = 0.f;
    }
    bf[ck].x = m0;
    bf[ck].y = m1;
  }
}

// ---------------------------------------------------------------------------
// Workspace layout (uint32 slots in d_ws):
//   [0..7]   minA bits   [8..15]  maxA bits
//   [16..23] minB bits   [24..31] maxB bits
//   [32..831] global histogram, int, 8 partitions x 100 bins
// All tracked values (means of states in [0,1]) are >= 0, so IEEE float
// ordering == unsigned-int bit ordering and atomicMin/Max on bits is exact.
// ---------------------------------------------------------------------------

__global__ void im_init_ws(unsigned int* mm, int* ghist) {
  int i = threadIdx.x;
  if (i < 8) {
    mm[i]      = 0x7f800000u;  // +inf
    mm[8 + i]  = 0u;
    mm[16 + i] = 0x7f800000u;
    mm[24 + i] = 0u;
  }
  for (int j = i; j < HSIZE; j += blockDim.x) ghist[j] = 0;
}

__global__ __launch_bounds__(256) void im_pass1_minmax(
    const float* __restrict__ states, const unsigned char* __restrict__ parts,
    unsigned int* mm, int numTiles, int totalWaves) {
  int tid  = threadIdx.x;
  int lane = tid & 31;
  int rl   = lane & 15;
  int h    = lane >> 4;
  int wave = (int)((blockIdx.x * blockDim.x + tid) >> 5);

  v2f bf[16];
  build_bfrag(parts, rl, h, bf);

  float ria = 0.f, rib = 0.f;
  if (rl < 8) {
    float cntA = 0.f;
    for (int d = 0; d < 64; ++d) cntA += parts[rl * 64 + d] ? 1.f : 0.f;
    ria = 1.f / cntA;
    rib = 1.f / (64.f - cntA);
  }

  // max accumulators start at 0 (values >= 0): safe even if a wave gets 0 tiles
  float mina = 3.4e38f, maxa = 0.f, minb = 3.4e38f, maxb = 0.f;

  for (long long t = wave; t < numTiles; t += totalWaves) {
    v8f acc = tile_sums(states, t, rl, h, bf);
#pragma unroll
    for (int r = 0; r < 8; ++r) {
      float v   = acc[r];
      float tot = __shfl(v, 8 + 16 * h, 32);  // column 8 = row total
      if (rl < 8) {
        float ma = v * ria;
        float mb = (tot - v) * rib;
        mina = fminf(mina, ma); maxa = fmaxf(maxa, ma);
        minb = fminf(minb, mb); maxb = fmaxf(maxb, mb);
      }
    }
  }

  if (rl < 8) {
    atomicMin(&mm[rl],      __float_as_uint(mina));
    atomicMax(&mm[8 + rl],  __float_as_uint(maxa));
    atomicMin(&mm[16 + rl], __float_as_uint(minb));
    atomicMax(&mm[24 + rl], __float_as_uint(maxb));
  }
}

__global__ __launch_bounds__(256) void im_pass2_hist(
    const float* __restrict__ states, const unsigned char* __restrict__ parts,
    const unsigned int* __restrict__ mm, int* ghist, int numTiles, int totalWaves) {
  __shared__ int hist[HSIZE];
  for (int i = threadIdx.x; i < HSIZE; i += blockDim.x) hist[i] = 0;
  __syncthreads();

  int tid  = threadIdx.x;
  int lane = tid & 31;
  int rl   = lane & 15;
  int h    = lane >> 4;
  int wave = (int)((blockIdx.x * blockDim.x + tid) >> 5);

  v2f bf[16];
  build_bfrag(parts, rl, h, bf);

  float ria = 0.f, rib = 0.f, mina = 0.f, minb = 0.f, ra = 0.f, rb = 0.f;
  if (rl < 8) {
    float cntA = 0.f;
    for (int d = 0; d < 64; ++d) cntA += parts[rl * 64 + d] ? 1.f : 0.f;
    ria = 1.f / cntA;
    rib = 1.f / (64.f - cntA);
    mina = __uint_as_float(mm[rl]);
    float maxa = __uint_as_float(mm[8 + rl]);
    minb = __uint_as_float(mm[16 + rl]);
    float maxb = __uint_as_float(mm[24 + rl]);
    ra = 1.f / (maxa - mina + 1e-6f);   // matches EPS_NORM
    rb = 1.f / (maxb - minb + 1e-6f);
  }

  for (long long t = wave; t < numTiles; t += totalWaves) {
    v8f acc = tile_sums(states, t, rl, h, bf);
#pragma unroll
    for (int r = 0; r < 8; ++r) {
      float v   = acc[r];
      float tot = __shfl(v, 8 + 16 * h, 32);
      if (rl < 8) {
        float sa = (v * ria - mina) * ra;
        float sb = ((tot - v) * rib - minb) * rb;
        int xb = (int)(sa * (float)NUM_BINS);
        int yb = (int)(sb * (float)NUM_BINS);
        xb = xb < 0 ? 0 : (xb > NUM_BINS - 1 ? NUM_BINS - 1 : xb);
        yb = yb < 0 ? 0 : (yb > NUM_BINS - 1 ? NUM_BINS - 1 : yb);
        atomicAdd(&hist[rl * NUM_BINS * NUM_BINS + xb * NUM_BINS + yb], 1);
      }
    }
  }

  __syncthreads();
  for (int i = threadIdx.x; i < HSIZE; i += blockDim.x) {
    int c = hist[i];
    if (c) atomicAdd(&ghist[i], c);
  }
}

__global__ void im_mi(const int* __restrict__ ghist, float* __restrict__ out) {
  __shared__ float mis[8];
  int p = threadIdx.x;
  if (p < 8) {
    float j[NUM_BINS * NUM_BINS];
    float tot = 0.f;
    for (int i = 0; i < NUM_BINS * NUM_BINS; ++i) {
      j[i] = (float)ghist[p * NUM_BINS * NUM_BINS + i];
      tot += j[i];
    }
    float inv = 1.f / (tot + 1e-10f);    // EPS_MI
    float px[NUM_BINS], py[NUM_BINS];
    for (int i = 0; i < NUM_BINS; ++i) { px[i] = 0.f; py[i] = 0.f; }
    for (int x = 0; x < NUM_BINS; ++x)
      for (int y = 0; y < NUM_BINS; ++y) {
        float jj = j[x * NUM_BINS + y] * inv;
        px[x] += jj;
        py[y] += jj;
      }
    float mi = 0.f;
    for (int x = 0; x < NUM_BINS; ++x)
      for (int y = 0; y < NUM_BINS; ++y) {
        float jj = j[x * NUM_BINS + y] * inv;
        mi += jj * __logf((jj + 1e-10f) / (px[x] * py[y] + 1e-10f));
      }
    float m = fmaxf(mi, 0.f);
    mis[p] = m;
    out[1 + p] = m;
  }
  __syncthreads();
  if (p == 0) {
    float mn = mis[0];
    for (int q = 1; q < 8; ++q) mn = fminf(mn, mis[q]);
    out[0] = mn;
  }
}

extern "C" void kernel_launch(void* const* d_in, const int* in_sizes, int n_in,
                              void* d_out, int out_size, void* d_ws, size_t ws_size,
                              hipStream_t stream) {
  const float* states        = (const float*)d_in[0];
  const unsigned char* parts = (const unsigned char*)d_in[1];  // jnp bool, 1B/elem
  float* out = (float*)d_out;

  int T = in_sizes[0] / 64;         // 2,000,000 (divisible by 16)
  int numTiles = T / 16;

  unsigned int* mm = (unsigned int*)d_ws;
  int* ghist = (int*)d_ws + 32;

  const int blocks = 1024, threads = 256;
  const int totalWaves = blocks * threads / 32;

  im_init_ws<<<1, 1024, 0, stream>>>(mm, ghist);
  im_pass1_minmax<<<blocks, threads, 0, stream>>>(states, parts, mm, numTiles, totalWaves);
  im_pass2_hist<<<blocks, threads, 0, stream>>>(states, parts, mm, ghist, numTiles, totalWaves);
  im_mi<<<1, 32, 0, stream>>>(ghist, out);
}